// HierarchicalGNN_1443109011558
// MI455X (gfx1250) — compile-verified
//
#include <hip/hip_runtime.h>
#include <hip/hip_bf16.h>

typedef __attribute__((ext_vector_type(2))) float v2f;
typedef __attribute__((ext_vector_type(8))) float v8f;

static inline unsigned cdiv(unsigned a, unsigned b) { return (a + b - 1) / b; }

// ---------------- elementwise / scatter helpers ----------------

__global__ void zero_f32(float* __restrict__ p, int n) {
    int i = blockIdx.x * blockDim.x + threadIdx.x;
    if (i < n) p[i] = 0.0f;
}

__global__ void deg_scatter(const int* __restrict__ dst, float* __restrict__ deg, int E) {
    int e = blockIdx.x * blockDim.x + threadIdx.x;
    if (e < E) atomicAdd(&deg[dst[e]], 1.0f);
}

__global__ void deg_to_dinv(float* __restrict__ deg, int n) {
    int i = blockIdx.x * blockDim.x + threadIdx.x;
    if (i < n) deg[i] = rsqrtf(deg[i] + 1.0f);   // +1 for self loop
}

// agg[n,f] = xw[n,f] * dinv[n]^2 + bias[f]   (self-loop contribution + bias)
__global__ void selfloop_init(const float* __restrict__ xw, const float* __restrict__ dinv,
                              const float* __restrict__ bias, float* __restrict__ agg,
                              int NF, int fshift) {
    int i = blockIdx.x * blockDim.x + threadIdx.x;
    if (i >= NF) return;
    int n = i >> fshift;
    int f = i & ((1 << fshift) - 1);
    float di = dinv[n];
    agg[i] = xw[i] * di * di + bias[f];
}

// per edge e, feature chunk c: agg[dst[e], c..c+3] += xw[src[e], c..c+3] * dinv[s]*dinv[d]
__global__ void edge_scatter(const float* __restrict__ xw, const int* __restrict__ src,
                             const int* __restrict__ dst, const float* __restrict__ dinv,
                             float* __restrict__ agg, int F, int E) {
    int e = blockIdx.x * blockDim.x + threadIdx.x;
    if (e >= E) return;
    int c = blockIdx.y * 4;
    int s = src[e], d = dst[e];
    float coef = dinv[s] * dinv[d];
    const float4 v = *(const float4*)(xw + (size_t)s * F + c);
    float* o = agg + (size_t)d * F + c;
    atomicAdd(o + 0, v.x * coef);
    atomicAdd(o + 1, v.y * coef);
    atomicAdd(o + 2, v.z * coef);
    atomicAdd(o + 3, v.w * coef);
}

__global__ void pool_scatter(const float* __restrict__ h, const int* __restrict__ batch,
                             float* __restrict__ sums, int N, int F) {
    int n = blockIdx.x * blockDim.x + threadIdx.x;
    if (n >= N) return;
    int c = blockIdx.y * 4;
    int g = batch[n];
    const float4 v = *(const float4*)(h + (size_t)n * F + c);
    float* o = sums + (size_t)g * F + c;
    atomicAdd(o + 0, v.x);
    atomicAdd(o + 1, v.y);
    atomicAdd(o + 2, v.z);
    atomicAdd(o + 3, v.w);
}

__global__ void cnt_scatter(const int* __restrict__ batch, float* __restrict__ cnts, int N) {
    int n = blockIdx.x * blockDim.x + threadIdx.x;
    if (n < N) atomicAdd(&cnts[batch[n]], 1.0f);
}

__global__ void pool_div(const float* __restrict__ sums, const float* __restrict__ cnts,
                         float* __restrict__ pooled, int GF, int fshift) {
    int i = blockIdx.x * blockDim.x + threadIdx.x;
    if (i < GF) pooled[i] = sums[i] / fmaxf(cnts[i >> fshift], 1.0f);
}

// ---------------- fp32 WMMA GEMM: C[M,N] = A[M,K] @ B[K,N] ----------------
// One wave computes a 16x64 strip (4 accumulators) via V_WMMA_F32_16X16X4_F32.
// The A fragment is loaded once per K-step and feeds 4 WMMAs (4 wmma / 10 loads).
// Requires M % 16 == 0 and N % 64 == 0.
// Layouts per CDNA5 ISA 7.12.2:
//   A 16x4 : lanes0-15 M=lane hold K=k,k+1 in v0,v1 ; lanes16-31 hold K=k+2,k+3
//   B 4x16 : lanes0-15 N=lane hold K=k,k+1 ; lanes16-31 K=k+2,k+3
//   D 16x16: VGPR r -> row r (lanes0-15, N=lane) / row r+8 (lanes16-31, N=lane-16)
template <bool RELU_A>
__global__ __launch_bounds__(256)
void gemm_wmma(const float* __restrict__ A, const float* __restrict__ B,
               float* __restrict__ C, int M, int N, int K) {
    const int lane = threadIdx.x & 31;
    const int wave = threadIdx.x >> 5;
    const int tiles_n = N >> 6;                 // strips of 64 columns
    int tile = blockIdx.x * 8 + wave;           // wave-uniform
    int tm = tile / tiles_n;
    int tn = tile - tm * tiles_n;
    if (tm * 16 >= M) return;                   // wave-uniform exit, EXEC stays full
    const int half = lane >> 4;
    const int l15  = lane & 15;
    const int row0 = tm * 16, col0 = tn * 64;
    const float* Ap = A + (size_t)(row0 + l15) * K + 2 * half;
    const float* Bp = B + (size_t)(2 * half) * N + col0 + l15;
    v8f c0 = {}, c1 = {}, c2 = {}, c3 = {};
    for (int k = 0; k < K; k += 4) {
        v2f a;
        a.x = Ap[0];
        a.y = Ap[1];
        if (RELU_A) { a.x = fmaxf(a.x, 0.0f); a.y = fmaxf(a.y, 0.0f); }
        v2f b0, b1, b2, b3;
        b0.x = Bp[0];       b0.y = Bp[N];
        b1.x = Bp[16];      b1.y = Bp[16 + N];
        b2.x = Bp[32];      b2.y = Bp[32 + N];
        b3.x = Bp[48];      b3.y = Bp[48 + N];
        c0 = __builtin_amdgcn_wmma_f32_16x16x4_f32(false, a, false, b0, (short)0, c0, false, false);
        c1 = __builtin_amdgcn_wmma_f32_16x16x4_f32(false, a, false, b1, (short)0, c1, false, false);
        c2 = __builtin_amdgcn_wmma_f32_16x16x4_f32(false, a, false, b2, (short)0, c2, false, false);
        c3 = __builtin_amdgcn_wmma_f32_16x16x4_f32(false, a, false, b3, (short)0, c3, false, false);
        Ap += 4;
        Bp += (size_t)4 * N;
    }
    float* Cp = C + (size_t)(row0 + 8 * half) * N + col0 + l15;
#pragma unroll
    for (int r = 0; r < 8; ++r) {
        Cp[(size_t)r * N + 0]  = c0[r];
        Cp[(size_t)r * N + 16] = c1[r];
        Cp[(size_t)r * N + 32] = c2[r];
        Cp[(size_t)r * N + 48] = c3[r];
    }
}

// Head GEMM with bias, narrow N (vn <= 16, padded to one 16-wide WMMA tile).
// Invalid lanes (l15 >= vn) clamp their B column to vn-1: their WMMA output
// columns are garbage-by-duplication but never stored, so loads stay
// unconditional (no exec save/restore in the hot loop).
// blockIdx.y = group index g: B += g*K*vn, bias += g*vn, out += g*M*vn.
__global__ __launch_bounds__(256)
void head_gemm(const float* __restrict__ A, const float* __restrict__ B,
               const float* __restrict__ bias, float* __restrict__ out,
               int M, int K, int vn) {
    int g = blockIdx.y;
    B    += (size_t)g * K * vn;
    bias += (size_t)g * vn;
    out  += (size_t)g * M * vn;
    const int lane = threadIdx.x & 31;
    const int wave = threadIdx.x >> 5;
    int tm = blockIdx.x * 8 + wave;
    if (tm * 16 >= M) return;                   // wave-uniform
    const int half = lane >> 4;
    const int l15  = lane & 15;
    const int row0 = tm * 16;
    const bool valid = (l15 < vn);
    const int bc = valid ? l15 : (vn - 1);      // clamped, always in-bounds
    const float* Ap = A + (size_t)(row0 + l15) * K + 2 * half;
    const float* Bp = B + (size_t)(2 * half) * vn + bc;
    v8f c = {};
    for (int k = 0; k < K; k += 4) {
        v2f a, b;
        a.x = Ap[0];
        a.y = Ap[1];
        b.x = Bp[0];
        b.y = Bp[vn];
        c = __builtin_amdgcn_wmma_f32_16x16x4_f32(false, a, false, b,
                                                  (short)0, c, false, false);
        Ap += 4;
        Bp += (size_t)4 * vn;
    }
    if (valid) {
        float bv = bias[l15];
#pragma unroll
        for (int r = 0; r < 8; ++r)
            out[(size_t)(row0 + 8 * half + r) * vn + l15] = c[r] + bv;
    }
}

// ---------------- launcher ----------------

extern "C" void kernel_launch(void* const* d_in, const int* in_sizes, int n_in,
                              void* d_out, int out_size, void* d_ws, size_t ws_size,
                              hipStream_t stream) {
    const float* x    = (const float*)d_in[0];
    const int*   edge = (const int*)d_in[1];
    const int*   batch= (const int*)d_in[2];
    const float* W1   = (const float*)d_in[3];
    const float* b1   = (const float*)d_in[4];
    const float* W2   = (const float*)d_in[5];
    const float* b2   = (const float*)d_in[6];
    const float* Wg   = (const float*)d_in[7];
    const float* bg   = (const float*)d_in[8];
    const float* Wf   = (const float*)d_in[9];
    const float* bf   = (const float*)d_in[10];

    const int N   = in_sizes[2];                 // 100000 nodes
    const int E   = in_sizes[1] / 2;             // 1.6M edges
    const int INF = in_sizes[0] / N;             // 128
    const int H1  = in_sizes[3] / INF;           // 128
    const int H2  = in_sizes[5] / H1;            // 256
    const int NG  = in_sizes[8];                 // 16 groups
    const int NF  = in_sizes[10] / (NG * H2);    // 10 families
    const int G   = out_size / (NG + NG * NF);   // 512 graphs
    const int s1  = 31 - __builtin_clz((unsigned)H1);  // log2(H1)
    const int s2  = 31 - __builtin_clz((unsigned)H2);  // log2(H2)

    const int* src = edge;
    const int* dst = edge + E;

    // workspace carving (256B aligned)
    char* w = (char*)d_ws;
    auto carve = [&](size_t bytes) -> float* {
        float* p = (float*)w;
        w += (bytes + 255) & ~(size_t)255;
        return p;
    };
    float* dinv   = carve((size_t)N * 4);
    float* xw1    = carve((size_t)N * H1 * 4);
    float* agg1   = carve((size_t)N * H1 * 4);
    float* xw2    = carve((size_t)N * H2 * 4);
    float* agg2   = carve((size_t)N * H2 * 4);
    float* sums   = carve((size_t)G * H2 * 4);
    float* cnts   = carve((size_t)G * 4);
    float* pooled = carve((size_t)G * H2 * 4);
    (void)ws_size;

    const int B = 256;

    // 1) degrees -> dinv = rsqrt(deg+1)
    zero_f32<<<cdiv(N, B), B, 0, stream>>>(dinv, N);
    deg_scatter<<<cdiv(E, B), B, 0, stream>>>(dst, dinv, E);
    deg_to_dinv<<<cdiv(N, B), B, 0, stream>>>(dinv, N);

    // 2) xw1 = x @ W1   [N,H1]
    {
        unsigned tiles = (unsigned)(N / 16) * (unsigned)(H1 / 64);
        gemm_wmma<false><<<cdiv(tiles, 8), B, 0, stream>>>(x, W1, xw1, N, H1, INF);
    }
    // 3) agg1 = xw1*dinv^2 + b1 ; scatter-add messages
    selfloop_init<<<cdiv((unsigned)N * H1, B), B, 0, stream>>>(xw1, dinv, b1, agg1, N * H1, s1);
    {
        dim3 grid(cdiv(E, B), H1 / 4);
        edge_scatter<<<grid, B, 0, stream>>>(xw1, src, dst, dinv, agg1, H1, E);
    }

    // 4) xw2 = relu(agg1) @ W2   [N,H2]   (relu fused into A-operand load)
    {
        unsigned tiles = (unsigned)(N / 16) * (unsigned)(H2 / 64);
        gemm_wmma<true><<<cdiv(tiles, 8), B, 0, stream>>>(agg1, W2, xw2, N, H2, H1);
    }
    // 5) agg2 = xw2*dinv^2 + b2 ; scatter-add messages
    selfloop_init<<<cdiv((unsigned)N * H2, B), B, 0, stream>>>(xw2, dinv, b2, agg2, N * H2, s2);
    {
        dim3 grid(cdiv(E, B), H2 / 4);
        edge_scatter<<<grid, B, 0, stream>>>(xw2, src, dst, dinv, agg2, H2, E);
    }

    // 6) mean pool over graphs
    zero_f32<<<cdiv((unsigned)G * H2, B), B, 0, stream>>>(sums, G * H2);
    zero_f32<<<cdiv(G, B), B, 0, stream>>>(cnts, G);
    {
        dim3 grid(cdiv(N, B), H2 / 4);
        pool_scatter<<<grid, B, 0, stream>>>(agg2, batch, sums, N, H2);
    }
    cnt_scatter<<<cdiv(N, B), B, 0, stream>>>(batch, cnts, N);
    pool_div<<<cdiv((unsigned)G * H2, B), B, 0, stream>>>(sums, cnts, pooled, G * H2, s2);

    // 7) heads
    float* out_group  = (float*)d_out;                  // [G, NG]
    float* out_family = (float*)d_out + (size_t)G * NG; // [NG, G, NF]
    {
        dim3 grid(cdiv(G / 16, 8), 1);
        head_gemm<<<grid, B, 0, stream>>>(pooled, Wg, bg, out_group, G, H2, NG);
    }
    {
        dim3 grid(cdiv(G / 16, 8), NG);
        head_gemm<<<grid, B, 0, stream>>>(pooled, Wf, bf, out_family, G, H2, NF);
    }
}